// WindowedScaledDotProductAttention_36747740184644
// MI455X (gfx1250) — compile-verified
//
#include <hip/hip_runtime.h>

typedef float v2f __attribute__((ext_vector_type(2)));
typedef float v4f __attribute__((ext_vector_type(4)));
typedef float v8f __attribute__((ext_vector_type(8)));

static constexpr int   kHeads        = 16;
static constexpr int   kSeq          = 4096;
static constexpr int   kDim          = 64;
static constexpr int   kHalfWin      = 128;
static constexpr float kNegInf       = -1000000000.0f;
static constexpr float kInvTemp      = 0.125f;     // 1/sqrt(64)
static constexpr int   kRowsPerWave  = 16;
static constexpr int   kWavesPerBlk  = 2;
static constexpr int   kLdsStride    = 276;        // 272 cols padded; 276%64==20 -> conflict-free

__global__ __launch_bounds__(kWavesPerBlk * 32)
void wattn16(const float* __restrict__ Q, const float* __restrict__ K,
             const float* __restrict__ V, float* __restrict__ Out,
             float* __restrict__ Attn)
{
    __shared__ float smem[kWavesPerBlk][kRowsPerWave * kLdsStride];

    const int lane = threadIdx.x & 31;
    const int wave = threadIdx.x >> 5;
    const int nl   = lane & 15;   // row (A) / col (B,C,D) sub-index
    const int g    = lane >> 4;   // K-group for A/B frags; row-half for C/D

    const int tileId = blockIdx.x * kWavesPerBlk + wave;
    const int head   = tileId >> 8;           // 256 row-tiles per head
    const int r0     = (tileId & 255) << 4;   // first query row of this tile

    const float* Qh = Q + (size_t)head * kSeq * kDim;
    const float* Kh = K + (size_t)head * kSeq * kDim;
    const float* Vh = V + (size_t)head * kSeq * kDim;
    float*       Oh = Out  + (size_t)head * kSeq * kDim;
    float*       Ah = Attn + (size_t)head * kSeq * kSeq;

    // window column-tile range (r0-128 is a multiple of 16, so >>4 is exact)
    int ct0 = (r0 - kHalfWin) >> 4;                 if (ct0 < 0) ct0 = 0;
    int ct1 = (r0 + kRowsPerWave - 1 + kHalfWin) >> 4;
    if (ct1 > (kSeq >> 4) - 1) ct1 = (kSeq >> 4) - 1;
    const int ntiles = ct1 - ct0 + 1;               // <= 17
    const int cstart = ct0 << 4;
    const int ncols  = ntiles << 4;                 // <= 272

    float* P = &smem[wave][0];

    // ---------------- Phase 1: S = (Q K^T) * invTemp, masked, -> LDS -------
    // A-fragment of Q tile (16 K-steps of 4): a[k] = Q[r0+nl][4k+2g .. +1]
    v2f aQ[16];
    {
        const float* qrow = Qh + (size_t)(r0 + nl) * kDim + 2 * g;
#pragma unroll
        for (int k = 0; k < 16; ++k)
            aQ[k] = *(const v2f*)(qrow + 4 * k);
    }

    for (int t = 0; t < ntiles; ++t) {
        const int c0 = cstart + (t << 4);
        const float* krow = Kh + (size_t)(c0 + nl) * kDim + 2 * g;
        v8f c = {0.f, 0.f, 0.f, 0.f, 0.f, 0.f, 0.f, 0.f};
#pragma unroll
        for (int k = 0; k < 16; ++k) {
            v2f b = *(const v2f*)(krow + 4 * k);
            c = __builtin_amdgcn_wmma_f32_16x16x4_f32(false, aQ[k], false, b,
                                                      (short)0, c, false, false);
        }
        const int col = c0 + nl;
#pragma unroll
        for (int r = 0; r < 8; ++r) {
            const int m  = r + 8 * g;       // C/D layout: row m in VGPR r, half g
            const int qr = r0 + m;
            int dlt = qr - col; if (dlt < 0) dlt = -dlt;
            const float s = (dlt <= kHalfWin) ? c[r] * kInvTemp : kNegInf;
            P[m * kLdsStride + (t << 4) + nl] = s;
        }
    }
    __syncthreads();

    // ---------------- Phase 2: row softmax in LDS, write attn rows ---------
    if (lane < 16) {
        float* prow = P + lane * kLdsStride;
        float mx = kNegInf;
        for (int i = 0; i < ncols; i += 4) {
            v4f q4 = *(const v4f*)(prow + i);
            mx = fmaxf(mx, fmaxf(fmaxf(q4.x, q4.y), fmaxf(q4.z, q4.w)));
        }
        float sum = 0.f;
        for (int i = 0; i < ncols; i += 4) {
            v4f q4 = *(const v4f*)(prow + i);
            q4.x = __expf(q4.x - mx);  q4.y = __expf(q4.y - mx);
            q4.z = __expf(q4.z - mx);  q4.w = __expf(q4.w - mx);
            sum += (q4.x + q4.y) + (q4.z + q4.w);
            *(v4f*)(prow + i) = q4;
        }
        const float inv = 1.0f / sum;
        for (int i = 0; i < ncols; i += 4) {
            v4f q4 = *(const v4f*)(prow + i);
            q4.x *= inv; q4.y *= inv; q4.z *= inv; q4.w *= inv;
            *(v4f*)(prow + i) = q4;
        }
    }
    __syncthreads();

    // full 4096-col attn rows: probs inside [cstart, cstart+ncols), zeros outside
    const v4f z4 = {0.f, 0.f, 0.f, 0.f};
    for (int r = 0; r < kRowsPerWave; ++r) {
        float*       arow = Ah + (size_t)(r0 + r) * kSeq;
        const float* prow = P + r * kLdsStride;
        for (int i = lane; i < (kSeq >> 2); i += 32) {
            const int col = i << 2;
            const int o   = col - cstart;
            v4f val = z4;
            if ((unsigned)o < (unsigned)ncols) val = *(const v4f*)(prow + o);
            __builtin_nontemporal_store(val, (v4f*)(arow + col));
        }
    }

    // ---------------- Phase 3: O = P V via WMMA ----------------------------
    v8f acc[4];
#pragma unroll
    for (int n = 0; n < 4; ++n) acc[n] = (v8f){0.f,0.f,0.f,0.f,0.f,0.f,0.f,0.f};

    for (int t = 0; t < ntiles; ++t) {
        const int vrow0 = cstart + (t << 4);
        v2f aP[4];
#pragma unroll
        for (int k = 0; k < 4; ++k) {
            const float* pp = P + nl * kLdsStride + (t << 4) + 4 * k + 2 * g;
            aP[k] = *(const v2f*)pp;   // 8-byte aligned (all terms even)
        }
#pragma unroll
        for (int n = 0; n < 4; ++n) {
#pragma unroll
            for (int k = 0; k < 4; ++k) {
                const float* vp = Vh + (size_t)(vrow0 + 4 * k + 2 * g) * kDim
                                     + n * 16 + nl;
                v2f b = { vp[0], vp[kDim] };   // two consecutive V rows, same col
                acc[n] = __builtin_amdgcn_wmma_f32_16x16x4_f32(false, aP[k], false, b,
                                                               (short)0, acc[n],
                                                               false, false);
            }
        }
    }

#pragma unroll
    for (int n = 0; n < 4; ++n) {
#pragma unroll
        for (int r = 0; r < 8; ++r) {
            const int m = r + 8 * g;
            Oh[(size_t)(r0 + m) * kDim + n * 16 + nl] = acc[n][r];
        }
    }
}

extern "C" void kernel_launch(void* const* d_in, const int* in_sizes, int n_in,
                              void* d_out, int out_size, void* d_ws, size_t ws_size,
                              hipStream_t stream) {
    const float* q = (const float*)d_in[0];
    const float* k = (const float*)d_in[1];
    const float* v = (const float*)d_in[2];
    float* out  = (float*)d_out;
    float* attn = out + (size_t)kHeads * kSeq * kDim;   // tuple: (out, attn) concatenated

    const int blocks = kHeads * (kSeq / kRowsPerWave) / kWavesPerBlk;  // 2048
    hipLaunchKernelGGL(wattn16, dim3(blocks), dim3(kWavesPerBlk * 32), 0, stream,
                       q, k, v, out, attn);
}